// ForwardKinematicsURDF_19902878450322
// MI455X (gfx1250) — compile-verified
//
#include <hip/hip_runtime.h>
#include <stdint.h>

// Forward kinematics over B independent serial chains (parent[i] = i-1).
// Memory-bound: reads 56 MB, writes 120 MB -> ~7.6 us floor @ 23.3 TB/s.
// Thread-per-chain; double-buffered async global->LDS input DMA (ASYNCcnt)
// overlapped with compute; outputs staged in LDS and flushed with
// non-temporal b128 stores (write side is the bound; NT keeps the 56 MB
// input resident in the 192 MB L2). pos_i == gpos_i - xyz_0, so pos is free.
//
// WMMA is deliberately not used: the only matrix work is a *serial* chain of
// dependent 3x3 f32 products (45 FLOPs each). The smallest CDNA5 WMMA tile is
// 16x16; block-diagonal packing would waste >90% of the tile, need per-step
// cross-lane shuffles costing more than the 9 FMAs they replace, and the f32
// shape (16x16x4) has no batched-diagonal form. The CDNA5 feature that
// actually pays here is the async-to-LDS DMA path.

#define NODES 64
#define TB    64          // chains (threads) per block
#define CH    8           // nodes per chunk
#define NCHUNK (NODES / CH)
#define ASYNC_PER_CHUNK 26  // per-wave async instrs per chunk: 2 (x,b128) + 24 (off,b64)

// LDS row strides in floats
#define XROW 12           // 8  + 4  (16B-aligned b128 dsts)
#define OROW 50           // 48 + 2  (8B-aligned b64 dsts; gcd(50,64)=2 -> conflict-free reads)
#define PROW 24           // pos/gpos staging, multiple of 4 for b128 reads
#define RROW 72           // rot staging, multiple of 4 for b128 reads

typedef __attribute__((ext_vector_type(4))) float v4f;

__device__ __forceinline__ uint32_t lds_off32(const void* p) {
  // Low 32 bits of a generic address of an LDS object = LDS byte offset.
  return (uint32_t)(uintptr_t)p;
}

__device__ __forceinline__ void async_ld_b64(uint32_t lds, const void* gaddr) {
  asm volatile("global_load_async_to_lds_b64 %0, %1, off"
               :: "v"(lds), "v"(gaddr) : "memory");
}
__device__ __forceinline__ void async_ld_b128(uint32_t lds, const void* gaddr) {
  asm volatile("global_load_async_to_lds_b128 %0, %1, off"
               :: "v"(lds), "v"(gaddr) : "memory");
}
__device__ __forceinline__ void wait_async_prev() {   // <= ASYNC_PER_CHUNK outstanding
  asm volatile("s_wait_asynccnt 26" ::: "memory");
}
__device__ __forceinline__ void wait_async_all() {
  asm volatile("s_wait_asynccnt 0" ::: "memory");
}

__device__ __forceinline__ void issue_chunk_async(int t, int b0, int nb, int i0,
                                                  const float* __restrict__ x,
                                                  const float* __restrict__ off,
                                                  float* sx, float* soff) {
  // x slice: TB chains x CH floats; 32B-aligned runs -> 2 b128 per thread
  #pragma unroll
  for (int k = 0; k < 2; ++k) {
    int q  = t + k * TB;             // [0, 2*TB)
    int tb = q >> 1, p = q & 1;      // p: which float4 of the 8-float run
    int bb = b0 + tb; if (bb >= nb) bb = nb - 1;      // clamp src; row unused
    const float* src = x + ((size_t)bb * NODES + i0 + 4 * p);
    async_ld_b128(lds_off32(&sx[tb * XROW + 4 * p]), src);
  }
  // offset slice: TB chains x CH*6 floats; runs only 8B-aligned -> 24 b64 per thread
  #pragma unroll
  for (int k = 0; k < 24; ++k) {
    int q  = t + k * TB;
    int tb = q / 24, p = q % 24;
    int bb = b0 + tb; if (bb >= nb) bb = nb - 1;
    const float* src = off + (((size_t)bb * NODES + i0) * 6 + 2 * p);
    async_ld_b64(lds_off32(&soff[tb * OROW + 2 * p]), src);
  }
}

__global__ __launch_bounds__(TB)
void fk_chain_kernel(const float* __restrict__ x,
                     const int*   __restrict__ par,
                     const float* __restrict__ off,
                     float* __restrict__ out,
                     int nb) {
  __shared__ __align__(16) float s_x  [2][TB * XROW];
  __shared__ __align__(16) float s_off[2][TB * OROW];
  __shared__ __align__(16) float s_pos [TB * PROW];
  __shared__ __align__(16) float s_gpos[TB * PROW];
  __shared__ __align__(16) float s_rot [TB * RROW];

  const int t  = threadIdx.x;
  const int b0 = blockIdx.x * TB;

  const size_t tot = (size_t)nb * NODES;
  float* __restrict__ opos  = out;            // (nb*N,3)
  float* __restrict__ orot  = out + tot * 3;  // (nb*N,3,3)
  float* __restrict__ ogpos = out + tot * 12; // (nb*N,3)

  // per-chain running state (registers)
  float R00=0,R01=0,R02=0,R10=0,R11=0,R12=0,R20=0,R21=0,R22=0;
  float g0=0,g1=0,g2=0, p00=0,p01=0,p02=0;

  // prologue: DMA chunk 0 into buffer 0
  issue_chunk_async(t, b0, nb, 0, x, off, s_x[0], s_off[0]);

  for (int ck = 0; ck < NCHUNK; ++ck) {
    const int i0  = ck * CH;
    const int cur = ck & 1;

    // prefetch next chunk, then wait only for the current one (in-order done)
    if (ck + 1 < NCHUNK) {
      issue_chunk_async(t, b0, nb, i0 + CH, x, off, s_x[cur ^ 1], s_off[cur ^ 1]);
      wait_async_prev();
    } else {
      wait_async_all();
    }
    __syncthreads();                 // chunk ck landed for every wave

    // ---- serial chain over this chunk (all state in registers) ----
    for (int j = 0; j < CH; ++j) {
      const int i = i0 + j;
      const float th = s_x[cur][t * XROW + j];
      const float* o6 = &s_off[cur][t * OROW + j * 6];
      const float x0 = o6[0], x1 = o6[1], x2 = o6[2];
      const float rx = o6[3], ry = o6[4], rz = o6[5];

      float sz,cz,sy,cy,sx,cx,st,ct;
      __sincosf(rz,&sz,&cz); __sincosf(ry,&sy,&cy);
      __sincosf(rx,&sx,&cx); __sincosf(th,&st,&ct);

      // Rrpy = Rz(rz) @ Ry(ry) @ Rx(rx)
      const float r00=cz*cy, r01=cz*sy*sx - sz*cx, r02=cz*sy*cx + sz*sx;
      const float r10=sz*cy, r11=sz*sy*sx + cz*cx, r12=sz*sy*cx - cz*sx;
      const float r20=-sy,   r21=cy*sx,            r22=cy*cx;
      // local = Rrpy @ Rz(th)
      const float l00=r00*ct + r01*st, l01=r01*ct - r00*st, l02=r02;
      const float l10=r10*ct + r11*st, l11=r11*ct - r10*st, l12=r12;
      const float l20=r20*ct + r21*st, l21=r21*ct - r20*st, l22=r22;

      if (par[i] < 0) {              // root (uniform scalar load)
        R00=l00;R01=l01;R02=l02;R10=l10;R11=l11;R12=l12;R20=l20;R21=l21;R22=l22;
        g0=x0; g1=x1; g2=x2;
        p00=x0; p01=x1; p02=x2;
      } else {                       // chain: parent is previous node
        g0 += R00*x0 + R01*x1 + R02*x2;
        g1 += R10*x0 + R11*x1 + R12*x2;
        g2 += R20*x0 + R21*x1 + R22*x2;
        const float n00=R00*l00+R01*l10+R02*l20, n01=R00*l01+R01*l11+R02*l21, n02=R00*l02+R01*l12+R02*l22;
        const float n10=R10*l00+R11*l10+R12*l20, n11=R10*l01+R11*l11+R12*l21, n12=R10*l02+R11*l12+R12*l22;
        const float n20=R20*l00+R21*l10+R22*l20, n21=R20*l01+R21*l11+R22*l21, n22=R20*l02+R21*l12+R22*l22;
        R00=n00;R01=n01;R02=n02;R10=n10;R11=n11;R12=n12;R20=n20;R21=n21;R22=n22;
      }

      // stage results (<=4-way bank conflicts here are negligible vs HBM bound)
      float* sp = &s_pos [t * PROW + j * 3];
      float* sg = &s_gpos[t * PROW + j * 3];
      float* sr = &s_rot [t * RROW + j * 9];
      sp[0]=g0-p00; sp[1]=g1-p01; sp[2]=g2-p02;
      sg[0]=g0;     sg[1]=g1;     sg[2]=g2;
      sr[0]=R00; sr[1]=R01; sr[2]=R02;
      sr[3]=R10; sr[4]=R11; sr[5]=R12;
      sr[6]=R20; sr[7]=R21; sr[8]=R22;
    }
    __syncthreads();                 // staging complete; also gates reuse of
                                     // input buffer cur^1 by later prefetches

    // ---- cooperative non-temporal b128 stores (write side is the bound) ----
    // pos & gpos: per chain 24 floats (96B, 16B-aligned since i0 % 8 == 0)
    #pragma unroll
    for (int k = 0; k < 6; ++k) {
      int q  = t + k * TB;           // [0, TB*6)
      int tb = q / 6, r = q % 6;
      if (b0 + tb < nb) {
        size_t gi = ((size_t)(b0 + tb) * NODES + i0) * 3 + 4 * (size_t)r;
        __builtin_nontemporal_store(*(const v4f*)&s_pos [tb * PROW + 4 * r], (v4f*)(opos  + gi));
        __builtin_nontemporal_store(*(const v4f*)&s_gpos[tb * PROW + 4 * r], (v4f*)(ogpos + gi));
      }
    }
    // rot: per chain 72 floats (288B, 16B-aligned)
    #pragma unroll
    for (int k = 0; k < 18; ++k) {
      int q  = t + k * TB;           // [0, TB*18)
      int tb = q / 18, r = q % 18;
      if (b0 + tb < nb) {
        size_t gi = ((size_t)(b0 + tb) * NODES + i0) * 9 + 4 * (size_t)r;
        __builtin_nontemporal_store(*(const v4f*)&s_rot[tb * RROW + 4 * r], (v4f*)(orot + gi));
      }
    }
    // no barrier needed here: the staging barrier above already proved every
    // wave finished reading input buffer (cur^1) from the previous chunk
  }
}

extern "C" void kernel_launch(void* const* d_in, const int* in_sizes, int n_in,
                              void* d_out, int out_size, void* d_ws, size_t ws_size,
                              hipStream_t stream) {
  const float* x   = (const float*)d_in[0];
  const int*   par = (const int*)  d_in[1];
  const float* off = (const float*)d_in[2];
  (void)n_in; (void)out_size; (void)d_ws; (void)ws_size;

  const int nb   = in_sizes[0] / NODES;          // B*N / N
  const int grid = (nb + TB - 1) / TB;
  fk_chain_kernel<<<grid, TB, 0, stream>>>(x, par, off, (float*)d_out, nb);
}